// INSIDE_12867722018862
// MI455X (gfx1250) — compile-verified
//
#include <hip/hip_runtime.h>
#include <math.h>

typedef __attribute__((ext_vector_type(2))) float v2f;
typedef __attribute__((ext_vector_type(4))) float f4;
typedef __attribute__((ext_vector_type(8))) float v8f;

#define CDIM 64
#define BDIM 2
#define BC   (BDIM * CDIM)          // 128
#define SZ   64                     // X=Y=Z=64
#define NOUT (BDIM * CDIM * SZ * SZ * SZ)   // 33554432
#define EPS32 1.1920929e-07f

// d_ws float layout
#define WS_SCALE 0
#define WS_SHIFT 128
#define WS_GX    256
#define WS_GY    (256 + BC * SZ)
#define WS_GZ    (256 + 2 * BC * SZ)

__device__ __forceinline__ float sigmoidf_(float v) {
    return 1.0f / (1.0f + expf(-v));
}

// ---------------------------------------------------------------------------
// Setup: MLP -> WMMA heads GEMM -> gaussian tables + l2
// one block, 256 threads (8 wave32)
// ---------------------------------------------------------------------------
__global__ void gauss_setup_kernel(const float* __restrict__ tab,
                                   const float* __restrict__ w1, const float* __restrict__ b1,
                                   const float* __restrict__ w2, const float* __restrict__ b2,
                                   const float* __restrict__ wp, const float* __restrict__ bp,
                                   const float* __restrict__ wsm, const float* __restrict__ bs,
                                   float* __restrict__ wsf, float* __restrict__ out) {
    __shared__ float s_emb1[BDIM * 32];
    __shared__ float s_A[16 * 32];      // zero-padded 16x32 A matrix (rows 0,1 = emb2)
    __shared__ float s_scale[BC];
    __shared__ float s_shift[BC];
    __shared__ float s_mu[3][BC];
    __shared__ float s_sig[3][BC];
    __shared__ float s_red[5][BC];

    const int t = threadIdx.x;

    // --- zero-pad A rows 2..15 (rows 0,1 overwritten in phase 2)
    for (int idx = t; idx < 16 * 32; idx += 256) s_A[idx] = 0.0f;

    // --- phase 1: emb1 = tanh(tab @ w1^T + b1), shapes (2,6)x(6,32)
    if (t < BDIM * 32) {
        int b = t >> 5, i = t & 31;
        float acc = b1[i];
        for (int k = 0; k < 6; ++k) acc += tab[b * 6 + k] * w1[i * 6 + k];
        s_emb1[b * 32 + i] = tanhf(acc);
    }
    __syncthreads();
    // --- phase 2: emb2 = tanh(emb1 @ w2^T + b2), shapes (2,32)x(32,32)
    if (t < BDIM * 32) {
        int b = t >> 5, j = t & 31;
        float acc = b2[j];
        for (int k = 0; k < 32; ++k) acc += s_emb1[b * 32 + k] * w2[j * 32 + k];
        s_A[b * 32 + j] = tanhf(acc);
    }
    __syncthreads();

    // --- phase 3: heads GEMM via V_WMMA_F32_16X16X4_F32
    // D(16x16) = A(16x4) x B(4x16) + C, accumulated over K=32 (8 steps).
    // A rows 0..1 = emb2 batch rows, rows 2..15 zero (pre-padded in LDS).
    // ISA A layout (16x4 f32): lane L holds M=L&15; VGPR0/1 = K={0,1} for
    // lanes 0-15, K={2,3} for lanes 16-31 -> one ds_load_b64 per K-step.
    // B layout (4x16): same K split, N = lane&15 within tile.
    const int lane = t & 31;
    const int wave = t >> 5;
    const int l    = lane & 15;       // A row m / B col within tile
    const int half = lane >> 4;       // K sub-pair selector
    for (int ti = 0; ti < 4; ++ti) {
        const int tileN = wave * 4 + ti;         // 0..31
        const int n = tileN * 16 + l;            // 0..511
        const float* Wrow;
        float bias;
        if (n < 320) { Wrow = wp + n * 32;        bias = bp[n]; }
        else         { Wrow = wsm + (n - 320) * 32; bias = bs[n - 320]; }

        v8f acc = {};
        for (int k0 = 0; k0 < 32; k0 += 4) {
            const int ka = k0 + 2 * half;
            const v2f av = *(const v2f*)&s_A[l * 32 + ka];   // unconditional b64
            v2f bv;
            bv.x = Wrow[ka];
            bv.y = Wrow[ka + 1];
            acc = __builtin_amdgcn_wmma_f32_16x16x4_f32(
                false, av, false, bv, (short)0, acc, false, false);
        }
        // extract rows M=0 (c[0], lanes 0-15) and M=1 (c[1], lanes 0-15)
        if (lane < 16) {
            float v0 = acc[0] + bias;   // batch b=0, head index n
            float v1 = acc[1] + bias;   // batch b=1
            if (n < 64) {
                s_scale[n] = v0; s_scale[64 + n] = v1;
            } else if (n < 128) {
                int c = n - 64;
                s_shift[c] = v0; s_shift[64 + c] = v1;
            } else if (n < 320) {
                int m = n - 128, d = m >> 6, c = m & 63;
                s_mu[d][c] = tanhf(v0); s_mu[d][64 + c] = tanhf(v1);
            } else {
                int m = n - 320, d = m >> 6, c = m & 63;
                s_sig[d][c] = sigmoidf_(v0); s_sig[d][64 + c] = sigmoidf_(v1);
            }
        }
    }
    __syncthreads();

    // --- phase 4: per-(b,c) gaussian tables + l2 partials
    if (t < BC) {
        const int bc = t;
        const float sc = s_scale[bc];
        const float sh = s_shift[bc];
        wsf[WS_SCALE + bc] = sc;
        wsf[WS_SHIFT + bc] = sh;
        s_red[0][bc] = sc * sc;
        s_red[1][bc] = sh * sh;
        for (int d = 0; d < 3; ++d) {
            const float sg = s_sig[d][bc];
            s_red[2 + d][bc] = sg * sg;
            const float mu  = s_mu[d][bc] * (SZ * 0.5f) + (SZ * 0.5f - 0.5f);
            const float inv = 1.0f / (sg * 3.5f + EPS32);
            float* gp = wsf + WS_GX + d * (BC * SZ) + bc * SZ;
            for (int i = 0; i < SZ; ++i) {
                float dd = ((float)i - mu) * inv;
                gp[i] = expf(-0.5f * dd * dd);
            }
        }
    }
    __syncthreads();
    if (t == 0) {
        float l2 = 0.0f;
        for (int q = 0; q < 5; ++q) {
            float s = 0.0f;
            for (int i = 0; i < BC; ++i) s += s_red[q][i];
            l2 += sqrtf(s);
        }
        out[NOUT] = l2;
    }
}

// ---------------------------------------------------------------------------
// Apply: out = (1-scale)*x*gx*gy*gz + shift     (HBM bandwidth bound)
// one block per (b,c,x) plane: 64x64 = 4096 f32 = 1024 float4
// 256 threads, 4 float4 each, non-temporal B128 streaming
// ---------------------------------------------------------------------------
__global__ void gauss_apply_kernel(const float* __restrict__ x,
                                   const float* __restrict__ wsf,
                                   float* __restrict__ out) {
    __shared__ float s_gy[SZ];
    __shared__ float s_gz[SZ];

    const int t   = threadIdx.x;
    const int bcx = blockIdx.x;          // 0..8191
    const int xc  = bcx & 63;
    const int bc  = bcx >> 6;            // 0..127

    if (t < SZ)            s_gy[t]      = wsf[WS_GY + bc * SZ + t];
    else if (t < 2 * SZ)   s_gz[t - SZ] = wsf[WS_GZ + bc * SZ + (t - SZ)];
    __syncthreads();

    const float scale = wsf[WS_SCALE + bc];
    const float shift = wsf[WS_SHIFT + bc];
    const float gxv   = wsf[WS_GX + bc * SZ + xc];
    const float k1    = (1.0f - scale) * gxv;

    const f4* __restrict__ xv4  = (const f4*)x;
    f4* __restrict__       ov4  = (f4*)out;
    const f4* __restrict__ gz4  = (const f4*)s_gz;
    const long base4 = (long)bcx * 1024;   // plane base in float4 units

    #pragma unroll
    for (int r = 0; r < 4; ++r) {
        const int g  = t + r * 256;        // 0..1023
        const int y  = g >> 4;
        const int z4 = g & 15;
        const f4 xv = __builtin_nontemporal_load(&xv4[base4 + g]);
        const f4 gz = gz4[z4];
        const float a = k1 * s_gy[y];
        f4 o;
        o.x = a * gz.x * xv.x + shift;
        o.y = a * gz.y * xv.y + shift;
        o.z = a * gz.z * xv.z + shift;
        o.w = a * gz.w * xv.w + shift;
        __builtin_nontemporal_store(o, &ov4[base4 + g]);
    }
}

extern "C" void kernel_launch(void* const* d_in, const int* in_sizes, int n_in,
                              void* d_out, int out_size, void* d_ws, size_t ws_size,
                              hipStream_t stream) {
    const float* x   = (const float*)d_in[0];
    const float* tab = (const float*)d_in[1];
    const float* w1  = (const float*)d_in[2];
    const float* b1  = (const float*)d_in[3];
    const float* w2  = (const float*)d_in[4];
    const float* b2  = (const float*)d_in[5];
    const float* wp  = (const float*)d_in[6];
    const float* bp  = (const float*)d_in[7];
    const float* wsm = (const float*)d_in[8];
    const float* bs  = (const float*)d_in[9];
    float* out = (float*)d_out;
    float* wsf = (float*)d_ws;

    gauss_setup_kernel<<<1, 256, 0, stream>>>(tab, w1, b1, w2, b2, wp, bp, wsm, bs,
                                              wsf, out);
    gauss_apply_kernel<<<BDIM * CDIM * SZ, 256, 0, stream>>>(x, wsf, out);
}